// MoEGate_74457553043890
// MI455X (gfx1250) — compile-verified
//
#include <hip/hip_runtime.h>
#include <hip/hip_bf16.h>
#include <math.h>

typedef __attribute__((ext_vector_type(2))) float v2f;
typedef __attribute__((ext_vector_type(8))) float v8f;

#define HID 5120
#define NEXP 160
#define NGRP 8
#define GSZ 20          // experts per group
#define TOPKG 3
#define TOPK 6
#define RSCALE 16.0f
#define LDS_STRIDE 164  // 160 + pad; 8*164 mod 64 = 32 -> lane halves hit different banks
#define TOK_PER_BLK 64

// Block = 64 tokens x 160 experts. 10 waves; wave w owns a 32x32 output block:
//   M-pair = w/5 -> tokens [t0+32*(w/5), +32);  N-pair = w%5 -> experts [32*(w%5), +32)
// 2x2 register tiling: 4 loads feed 4 V_WMMA_F32_16X16X4_F32 per K-step (1 load/wmma).
__global__ __launch_bounds__(320) void moe_gate_kernel(const float* __restrict__ x,
                                                       const float* __restrict__ w,
                                                       float* __restrict__ out) {
    __shared__ float logits[TOK_PER_BLK * LDS_STRIDE];

    const int t0   = blockIdx.x * TOK_PER_BLK;
    const int wave = threadIdx.x >> 5;   // 0..9
    const int lane = threadIdx.x & 31;
    const int half = lane >> 4;          // K sub-pair selector per ISA fp32 A/B layout
    const int m    = lane & 15;          // M for A tiles, N for B tiles

    const int mbase = (wave / 5) * 32;   // 0 or 32 (token offset within block)
    const int e0    = (wave % 5) * 32;   // 0,32,...,128 (expert base)

    const float* a0 = x + (size_t)(t0 + mbase + m) * HID + 2 * half;
    const float* a1 = x + (size_t)(t0 + mbase + 16 + m) * HID + 2 * half;
    const float* b0 = w + (size_t)(e0 + m) * HID + 2 * half;
    const float* b1 = w + (size_t)(e0 + 16 + m) * HID + 2 * half;

    v8f acc00 = {}, acc01 = {}, acc10 = {}, acc11 = {};
#pragma unroll 4
    for (int k = 0; k < HID; k += 4) {
        v2f va0 = *(const v2f*)(a0 + k);
        v2f va1 = *(const v2f*)(a1 + k);
        v2f vb0 = *(const v2f*)(b0 + k);
        v2f vb1 = *(const v2f*)(b1 + k);
        acc00 = __builtin_amdgcn_wmma_f32_16x16x4_f32(false, va0, false, vb0, (short)0, acc00, false, false);
        acc01 = __builtin_amdgcn_wmma_f32_16x16x4_f32(false, va0, false, vb1, (short)0, acc01, false, false);
        acc10 = __builtin_amdgcn_wmma_f32_16x16x4_f32(false, va1, false, vb0, (short)0, acc10, false, false);
        acc11 = __builtin_amdgcn_wmma_f32_16x16x4_f32(false, va1, false, vb1, (short)0, acc11, false, false);
    }

    // C/D layout: VGPR i -> (M = i + 8*half, N = m) within each 16x16 tile
#pragma unroll
    for (int i = 0; i < 8; ++i) {
        const int r0 = mbase + i + 8 * half;
        logits[r0 * LDS_STRIDE + e0 + m]             = acc00[i];
        logits[r0 * LDS_STRIDE + e0 + 16 + m]        = acc01[i];
        logits[(r0 + 16) * LDS_STRIDE + e0 + m]      = acc10[i];
        logits[(r0 + 16) * LDS_STRIDE + e0 + 16 + m] = acc11[i];
    }
    __syncthreads();

    // ---- routing epilogue: one lane per token (lanes 0..63) ----
    if (threadIdx.x < TOK_PER_BLK) {
        const float* row = &logits[threadIdx.x * LDS_STRIDE];

        // per-group max logit and global max
        float gmax[NGRP];
        float M = -INFINITY;
#pragma unroll
        for (int g = 0; g < NGRP; ++g) {
            float mg = row[g * GSZ];
            for (int j = 1; j < GSZ; ++j) mg = fmaxf(mg, row[g * GSZ + j]);
            gmax[g] = mg;
            M = fmaxf(M, mg);
        }

        // softmax denominator over ALL experts
        float sum = 0.0f;
        for (int e = 0; e < NEXP; ++e) sum += expf(row[e] - M);

        // top-3 groups by max logit (monotonic == top-3 by max score); lowest-index tie-break
        unsigned sel = 0;
#pragma unroll
        for (int t = 0; t < TOPKG; ++t) {
            int bi = 0; float bv = -INFINITY;
            for (int g = 0; g < NGRP; ++g) {
                if (!((sel >> g) & 1u) && gmax[g] > bv) { bv = gmax[g]; bi = g; }
            }
            sel |= 1u << bi;
        }

        // top-6 logits within selected groups (descending; strict '>' keeps earlier index on ties)
        float top[TOPK];
#pragma unroll
        for (int i = 0; i < TOPK; ++i) top[i] = -INFINITY;
        for (int g = 0; g < NGRP; ++g) {
            if (!((sel >> g) & 1u)) continue;
            for (int j = 0; j < GSZ; ++j) {
                float v = row[g * GSZ + j];
                if (v > top[TOPK - 1]) {
                    int p = TOPK - 1;
                    while (p > 0 && v > top[p - 1]) { top[p] = top[p - 1]; --p; }
                    top[p] = v;
                }
            }
        }

        const float inv = RSCALE / sum;
        float* o = out + (size_t)(t0 + threadIdx.x) * TOPK;
#pragma unroll
        for (int i = 0; i < TOPK; ++i) o[i] = expf(top[i] - M) * inv;
    }
}

extern "C" void kernel_launch(void* const* d_in, const int* in_sizes, int n_in,
                              void* d_out, int out_size, void* d_ws, size_t ws_size,
                              hipStream_t stream) {
    const float* hidden = (const float*)d_in[0];  // [T, 5120] fp32
    const float* kern   = (const float*)d_in[1];  // [160, 5120] fp32
    float* out          = (float*)d_out;          // [T, 6] fp32

    const int T = in_sizes[0] / HID;              // 8192
    const int nblocks = T / TOK_PER_BLK;          // 128

    moe_gate_kernel<<<nblocks, 320, 0, stream>>>(hidden, kern, out);
}